// GramSchmidtLayer_33681133536122
// MI455X (gfx1250) — compile-verified
//
#include <hip/hip_runtime.h>
#include <hip/hip_bf16.h>

// CholeskyQR2 for batched 64x16 complex QR on gfx1250.
// One wave (32 lanes) per matrix. All matmuls via V_WMMA_F32_16X16X4_F32.

typedef __attribute__((ext_vector_type(2))) float v2f;
typedef __attribute__((ext_vector_type(8))) float v8f;

#define LDS_S 17   // padded row stride (floats) to avoid bank conflicts

__device__ __forceinline__ v8f wmma4(v2f a, v2f b, v8f c) {
  // D = A(16x4) * B(4x16) + C(16x16), fp32
  return __builtin_amdgcn_wmma_f32_16x16x4_f32(
      /*neg_a=*/false, a, /*neg_b=*/false, b,
      /*c_mod=*/(short)0, c, /*reuse_a=*/false, /*reuse_b=*/false);
}

__global__ __launch_bounds__(32)
void cholqr2_c64_kernel(const float* __restrict__ x, float* __restrict__ out) {
  __shared__ float sAr[64 * LDS_S], sAi[64 * LDS_S];   // A / Q planes
  __shared__ float sGr[16 * LDS_S], sGi[16 * LDS_S];   // Gram -> Cholesky L
  __shared__ float sWr[16 * LDS_S], sWi[16 * LDS_S];   // W = L^-1
  __shared__ float sRr[16 * LDS_S], sRi[16 * LDS_S];   // Rinv = W^H

  const int lane  = threadIdx.x;        // 0..31
  const int c     = lane & 15;          // column / row-in-tile index
  const int khalf = (lane >> 4) << 1;   // 0 for lanes 0-15, 2 for lanes 16-31
  const long b    = blockIdx.x;

  // ---- load A: [64][16] complex interleaved -> split re/im LDS planes ----
  const float2* src = (const float2*)(x + b * 2048);
  for (int t = lane; t < 1024; t += 32) {
    int m = t >> 4, r = t & 15;
    float2 v = src[t];
    sAr[m * LDS_S + r] = v.x;
    sAi[m * LDS_S + r] = v.y;
  }
  __syncthreads();

  for (int pass = 0; pass < 2; ++pass) {
    // ---- G = A^H A via WMMA (16 K-chunks of 4) ----
    v8f gr = {}, gi = {};
    for (int kk = 0; kk < 64; kk += 4) {
      const int k0 = kk + khalf;
      v2f fr, fi;
      fr.x = sAr[k0 * LDS_S + c];        fr.y = sAr[(k0 + 1) * LDS_S + c];
      fi.x = sAi[k0 * LDS_S + c];        fi.y = sAi[(k0 + 1) * LDS_S + c];
      v2f nfi = -fi;
      // Gr += Ar^T Ar + Ai^T Ai ; Gi += Ar^T Ai - Ai^T Ar
      gr = wmma4(fr, fr, gr);
      gr = wmma4(fi, fi, gr);
      gi = wmma4(fr, fi, gi);
      gi = wmma4(nfi, fr, gi);
    }
    // spill C/D fragment -> LDS: VGPR v holds M=v(+8 for hi lanes), N=c
#pragma unroll
    for (int v = 0; v < 8; ++v) {
      int i = v + ((lane >> 4) << 3);
      sGr[i * LDS_S + c] = gr[v];
      sGi[i * LDS_S + c] = gi[v];
    }
    __syncthreads();

    // ---- Cholesky: G = L L^H (in place, lower + real diag) ----
    for (int j = 0; j < 16; ++j) {
      float d    = sqrtf(fmaxf(sGr[j * LDS_S + j], 1e-30f));
      float invd = 1.0f / d;
      if (lane == j) { sGr[j * LDS_S + j] = d; sGi[j * LDS_S + j] = 0.0f; }
      if (lane > j && lane < 16) {
        sGr[lane * LDS_S + j] *= invd;
        sGi[lane * LDS_S + j] *= invd;
      }
      __syncthreads();
      if (lane > j && lane < 16) {
        float lr = sGr[lane * LDS_S + j], li = sGi[lane * LDS_S + j];
        for (int k = j + 1; k <= lane; ++k) {
          float mr = sGr[k * LDS_S + j], mi = sGi[k * LDS_S + j];
          sGr[lane * LDS_S + k] -= lr * mr + li * mi;   // L[i][j]*conj(L[k][j])
          sGi[lane * LDS_S + k] -= li * mr - lr * mi;
        }
      }
      __syncthreads();
    }

    // ---- W = L^-1 (forward substitution, lane j owns column j) ----
    if (lane < 16) {
      const int j = lane;
      for (int i = 0; i < j; ++i) { sWr[i * LDS_S + j] = 0.0f; sWi[i * LDS_S + j] = 0.0f; }
      sWr[j * LDS_S + j] = 1.0f / sGr[j * LDS_S + j];
      sWi[j * LDS_S + j] = 0.0f;
      for (int i = j + 1; i < 16; ++i) {
        float sr = 0.0f, si = 0.0f;
        for (int k = j; k < i; ++k) {
          float lr = sGr[i * LDS_S + k], li = sGi[i * LDS_S + k];
          float wr = sWr[k * LDS_S + j], wi = sWi[k * LDS_S + j];
          sr += lr * wr - li * wi;
          si += lr * wi + li * wr;
        }
        float inv = 1.0f / sGr[i * LDS_S + i];
        sWr[i * LDS_S + j] = -sr * inv;
        sWi[i * LDS_S + j] = -si * inv;
      }
    }
    __syncthreads();
    // Rinv = W^H : lane j writes row j (upper triangular)
    if (lane < 16) {
      const int j = lane;
      for (int i = 0; i < 16; ++i) {
        bool up = (i >= j);
        sRr[j * LDS_S + i] = up ?  sWr[i * LDS_S + j] : 0.0f;
        sRi[j * LDS_S + i] = up ? -sWi[i * LDS_S + j] : 0.0f;
      }
    }
    __syncthreads();

    // ---- Q = A * Rinv via WMMA, 4 row tiles of 16 ----
    for (int m0 = 0; m0 < 64; m0 += 16) {
      v8f qr = {}, qi = {};
      const int row = m0 + c;
      for (int kk = 0; kk < 16; kk += 4) {
        const int k0 = kk + khalf;
        v2f ar, ai, br, bi;
        ar.x = sAr[row * LDS_S + k0];      ar.y = sAr[row * LDS_S + k0 + 1];
        ai.x = sAi[row * LDS_S + k0];      ai.y = sAi[row * LDS_S + k0 + 1];
        br.x = sRr[k0 * LDS_S + c];        br.y = sRr[(k0 + 1) * LDS_S + c];
        bi.x = sRi[k0 * LDS_S + c];        bi.y = sRi[(k0 + 1) * LDS_S + c];
        v2f nai = -ai;
        // Qr += Ar*Rr - Ai*Ri ; Qi += Ar*Ri + Ai*Rr
        qr = wmma4(ar,  br, qr);
        qr = wmma4(nai, bi, qr);
        qi = wmma4(ar,  bi, qi);
        qi = wmma4(ai,  br, qi);
      }
#pragma unroll
      for (int v = 0; v < 8; ++v) {
        int i = m0 + v + ((lane >> 4) << 3);   // global row of this accumulator slot
        if (pass == 0) {
          sAr[i * LDS_S + c] = qr[v];
          sAi[i * LDS_S + c] = qi[v];
        } else {
          float2 o; o.x = qr[v]; o.y = qi[v];
          ((float2*)out)[b * 1024 + i * 16 + c] = o;
        }
      }
    }
    __syncthreads();
  }
}

extern "C" void kernel_launch(void* const* d_in, const int* in_sizes, int n_in,
                              void* d_out, int out_size, void* d_ws, size_t ws_size,
                              hipStream_t stream) {
  const float* x = (const float*)d_in[0];
  float* out = (float*)d_out;
  const int batch = in_sizes[0] / (64 * 16 * 2);   // 32768
  cholqr2_c64_kernel<<<batch, 32, 0, stream>>>(x, out);
}